// HybridKernelAutoLSTM_65481071395817
// MI455X (gfx1250) — compile-verified
//
#include <hip/hip_runtime.h>
#include <hip/hip_bf16.h>

// ---------------------------------------------------------------------------
// QLSTM on MI455X (gfx1250, wave32).
//
// Kernel 1 (scan): one lane per (batch element, gate). Wave = 8 elems x 4
//   gates. Per step: 48-FMA matvec, 4x v_cos_f32 (qlayer collapsed
//   analytically to products of cosines), tanh-based activations, gate
//   exchange via ds_bpermute, h/c update, coalesced store of h.
// Kernel 2 (projection): two v_wmma_f32_16x16x4_f32 per iteration compute
//   outs(16x4) @ w_tag^T(4x16) + b_tag for 32 rows; the 16x16 tiles are
//   transposed through per-wave LDS so each lane owns one full row and the
//   log_softmax runs entirely in-lane (no cross-lane reductions).
// ---------------------------------------------------------------------------

#define T_STEPS 512
#define BATCH   2048
#define DIN     8
#define HDIM    4
#define TAGS    16

typedef float v2f __attribute__((ext_vector_type(2)));
typedef float v8f __attribute__((ext_vector_type(8)));

__device__ __forceinline__ float lane_read_f(float v, int srcLane) {
  // wave32 cross-lane gather: Dst[lane] = src[srcLane]; index in bytes.
  return __int_as_float(
      __builtin_amdgcn_ds_bpermute(srcLane << 2, __float_as_int(v)));
}

// ---------------------------------------------------------------------------
// Kernel 1: sequential LSTM scan. grid = BATCH/8 blocks of 32 threads.
// ---------------------------------------------------------------------------
__global__ void qlstm_scan_kernel(const float* __restrict__ x,
                                  const float* __restrict__ w_gates,
                                  const float* __restrict__ b_gates,
                                  const float* __restrict__ rx_theta,
                                  float* __restrict__ outs) {
  const int lane = (int)threadIdx.x;      // 0..31
  const int g    = lane & 3;              // gate: 0=f 1=i 2=g(tanh) 3=o
  const int e    = lane >> 2;             // element within wave (0..7)
  const int b    = (int)blockIdx.x * 8 + e;
  const int base4 = lane & ~3;

  // Per-gate weights (4 outputs x 12 inputs), bias folded with rx_theta:
  // cos argument = W.comb + b_gates + rx_theta.
  float W[4][12];
#pragma unroll
  for (int k = 0; k < 4; ++k)
#pragma unroll
    for (int d = 0; d < 12; ++d)
      W[k][d] = w_gates[g * 48 + k * 12 + d];

  float bias[4];
#pragma unroll
  for (int k = 0; k < 4; ++k)
    bias[k] = b_gates[g * 4 + k] + rx_theta[g * 4 + k];

  // sigmoid(x) = 0.5 + 0.5*tanh(0.5x); gate 2 uses plain tanh.
  const float tsc = (g == 2) ? 1.0f : 0.5f;
  const float tof = (g == 2) ? 0.0f : 0.5f;

  float h[4] = {0.f, 0.f, 0.f, 0.f};
  float c[4] = {0.f, 0.f, 0.f, 0.f};

  const float4* x4 = (const float4*)x;

  for (int t = 0; t < T_STEPS; ++t) {
    const size_t row = (size_t)t * BATCH + (size_t)b;
    float4 xa = x4[row * 2 + 0];
    float4 xb = x4[row * 2 + 1];
    float xv[8] = {xa.x, xa.y, xa.z, xa.w, xb.x, xb.y, xb.z, xb.w};

    if (t + 8 < T_STEPS)  // global_prefetch_b8: pull future timestep lines
      __builtin_prefetch(x + (row + 8u * BATCH) * 8, 0, 1);

    // pre-activations -> C_k = cos(pre_k + theta_k)
    float C[4];
#pragma unroll
    for (int k = 0; k < 4; ++k) {
      float p = bias[k];
#pragma unroll
      for (int d = 0; d < 8; ++d) p = __builtin_fmaf(W[k][d], xv[d], p);
#pragma unroll
      for (int j = 0; j < 4; ++j) p = __builtin_fmaf(W[k][8 + j], h[j], p);
      C[k] = __cosf(p);
    }

    // Collapsed qlayer expvals (CNOT ring -> cosine products):
    float ev[4];
    const float c01 = C[0] * C[1];
    ev[1] = c01;
    ev[2] = c01 * C[2];
    ev[3] = ev[2] * C[3];
    ev[0] = C[1] * (C[2] * C[3]);

    float a[4];
#pragma unroll
    for (int k = 0; k < 4; ++k) a[k] = tof + tsc * tanhf(tsc * ev[k]);

    // Gather all four gate vectors from the 4 lanes of this element and
    // update carried state (each lane redundantly, stays consistent).
#pragma unroll
    for (int k = 0; k < 4; ++k) {
      float fk = lane_read_f(a[k], base4 + 0);
      float ik = lane_read_f(a[k], base4 + 1);
      float gk = lane_read_f(a[k], base4 + 2);
      float ok = lane_read_f(a[k], base4 + 3);
      c[k] = __builtin_fmaf(fk, c[k], ik * gk);
      h[k] = ok * tanhf(c[k]);
    }

    // Lane with gate g stores component h[g]: 32 contiguous floats per wave.
    outs[row * HDIM + g] = h[g];
  }
}

// ---------------------------------------------------------------------------
// Kernel 2: logits = outs @ w_tag^T + b_tag via 2x WMMA f32 16x16x4 per
//           iteration (32 rows), LDS transpose, in-lane log_softmax.
// ---------------------------------------------------------------------------
__global__ void logits_lsm_kernel(const float* __restrict__ outs,
                                  const float* __restrict__ w_tag,
                                  const float* __restrict__ b_tag,
                                  float* __restrict__ out,
                                  int numPairs) {
  // Per-wave private LDS: 2 tiles x 256 floats. 8 waves/block -> 16 KB.
  __shared__ __align__(16) float lds[8 * 512];

  const int lane  = (int)(threadIdx.x & 31u);
  const int wslot = (int)(threadIdx.x >> 5);
  float* wl = &lds[wslot * 512];

  const int n  = lane & 15;  // WMMA column (tag)
  const int hi = lane >> 4;  // K-half: lanes 0-15 -> k=0,1; 16-31 -> k=2,3
  const int wavesPerBlk = (int)(blockDim.x >> 5);
  const int wid = (int)blockIdx.x * wavesPerBlk + wslot;
  const int nw  = (int)gridDim.x * wavesPerBlk;

  // B fragment: B[k][n] = w_tag[n*4 + k]; lane holds k = 2*hi, 2*hi+1.
  const float2* wt2 = (const float2*)w_tag;
  float2 bv = wt2[n * 2 + hi];
  v2f bfrag = {bv.x, bv.y};
  const float bias = b_tag[n];

  const float2* a2 = (const float2*)outs;

  for (int pair = wid; pair < numPairs; pair += nw) {
    const size_t rowBase = (size_t)pair * 32;

    // A fragments for the two 16-row tiles (contiguous float2 per lane).
    float2 av0 = a2[(rowBase + (size_t)n) * 2 + hi];
    float2 av1 = a2[(rowBase + 16 + (size_t)n) * 2 + hi];
    v2f af0 = {av0.x, av0.y};
    v2f af1 = {av1.x, av1.y};

    v8f acc0, acc1;
#pragma unroll
    for (int j = 0; j < 8; ++j) { acc0[j] = bias; acc1[j] = bias; }

    acc0 = __builtin_amdgcn_wmma_f32_16x16x4_f32(false, af0, false, bfrag,
                                                 (short)0, acc0, false, false);
    acc1 = __builtin_amdgcn_wmma_f32_16x16x4_f32(false, af1, false, bfrag,
                                                 (short)0, acc1, false, false);

    // Scatter both tiles row-major into per-wave LDS:
    // lane holds rows hi*8+j (j=0..7), column n of each tile.
#pragma unroll
    for (int j = 0; j < 8; ++j) wl[(hi * 8 + j) * 16 + n] = acc0[j];
#pragma unroll
    for (int j = 0; j < 8; ++j) wl[256 + (hi * 8 + j) * 16 + n] = acc1[j];

    // Each lane now owns one full row of the 32: tile = lane>>4, row = lane&15.
    const float4* wl4 = (const float4*)wl;
    const int rb = (lane >> 4) * 64 + (lane & 15) * 4;
    float4 q0 = wl4[rb + 0];
    float4 q1 = wl4[rb + 1];
    float4 q2 = wl4[rb + 2];
    float4 q3 = wl4[rb + 3];
    float vv[16] = {q0.x, q0.y, q0.z, q0.w, q1.x, q1.y, q1.z, q1.w,
                    q2.x, q2.y, q2.z, q2.w, q3.x, q3.y, q3.z, q3.w};

    // In-lane log_softmax over the 16 tags.
    float m0 = fmaxf(vv[0], vv[1]), m1 = fmaxf(vv[2], vv[3]);
    float m2 = fmaxf(vv[4], vv[5]), m3 = fmaxf(vv[6], vv[7]);
    float m4 = fmaxf(vv[8], vv[9]), m5 = fmaxf(vv[10], vv[11]);
    float m6 = fmaxf(vv[12], vv[13]), m7 = fmaxf(vv[14], vv[15]);
    m0 = fmaxf(m0, m1); m2 = fmaxf(m2, m3);
    m4 = fmaxf(m4, m5); m6 = fmaxf(m6, m7);
    m0 = fmaxf(m0, m2); m4 = fmaxf(m4, m6);
    const float m = fmaxf(m0, m4);

    float s = 0.f;
#pragma unroll
    for (int j = 0; j < 16; ++j) s += __expf(vv[j] - m);
    const float off = m + __logf(s);

    float4 o0 = {vv[0] - off, vv[1] - off, vv[2] - off, vv[3] - off};
    float4 o1 = {vv[4] - off, vv[5] - off, vv[6] - off, vv[7] - off};
    float4 o2 = {vv[8] - off, vv[9] - off, vv[10] - off, vv[11] - off};
    float4 o3 = {vv[12] - off, vv[13] - off, vv[14] - off, vv[15] - off};

    float4* op = (float4*)(out + (rowBase + (size_t)lane) * TAGS);
    op[0] = o0; op[1] = o1; op[2] = o2; op[3] = o3;
  }
}

// ---------------------------------------------------------------------------
extern "C" void kernel_launch(void* const* d_in, const int* in_sizes, int n_in,
                              void* d_out, int out_size, void* d_ws,
                              size_t ws_size, hipStream_t stream) {
  const float* x        = (const float*)d_in[0];  // (T,B,D)
  const float* w_gates  = (const float*)d_in[1];  // (4,4,12)
  const float* b_gates  = (const float*)d_in[2];  // (4,4)
  const float* rx_theta = (const float*)d_in[3];  // (4,4)
  const float* w_tag    = (const float*)d_in[4];  // (16,4)
  const float* b_tag    = (const float*)d_in[5];  // (16,)

  float* outs = (float*)d_ws;  // (T,B,H) = 16 MiB scratch, fully overwritten

  // Scan: BATCH/8 single-wave workgroups (all comms intra-wave).
  qlstm_scan_kernel<<<BATCH / 8, 32, 0, stream>>>(x, w_gates, b_gates,
                                                  rx_theta, outs);

  // Projection + log_softmax: 32768 pairs of 16-row tiles, grid-stride/wave.
  const int numPairs = (T_STEPS * BATCH) / 32;
  logits_lsm_kernel<<<256, 256, 0, stream>>>(outs, w_tag, b_tag, (float*)d_out,
                                             numPairs);
}